// GraphConvolution_90460601189195
// MI455X (gfx1250) — compile-verified
//
#include <hip/hip_runtime.h>
#include <hip/hip_bf16.h>

typedef __attribute__((ext_vector_type(2))) float v2f;
typedef __attribute__((ext_vector_type(8))) float v8f;

#define IN_FEAT 256
#define UNITS   256
#define LDS_PITCH 260   // 16-row x-tile, padded to spread LDS banks (260 % 64 == 4)

// ---------------------------------------------------------------------------
// Dense projection h = x @ w with V_WMMA_F32_16X16X4_F32.
// Grid: one block per 16-row tile of x (N_NODES is an exact multiple of 16).
// Block: 256 threads = 8 wave32s; wave w computes output columns [w*32, w*32+32).
// ---------------------------------------------------------------------------
__global__ void __launch_bounds__(256) gcn_gemm_wmma_f32(
    const float* __restrict__ x, const float* __restrict__ w_mat,
    float* __restrict__ h)
{
    __shared__ float atile[16 * LDS_PITCH];

    const int tid = threadIdx.x;
    const int m0  = blockIdx.x * 16;

    // Cooperatively stage the 16x256 fp32 x-tile into LDS (16 floats/thread).
    {
        const int row = tid >> 4;            // 0..15
        const int cb  = (tid & 15) * 16;     // 0..240 step 16
        const float4* src = (const float4*)(x + (size_t)(m0 + row) * IN_FEAT + cb);
        float4* dst = (float4*)(atile + row * LDS_PITCH + cb);  // 1040B row pitch, 16B aligned
        dst[0] = src[0]; dst[1] = src[1]; dst[2] = src[2]; dst[3] = src[3];
    }
    __syncthreads();

    const int lane = tid & 31;
    const int wid  = tid >> 5;
    const int n0   = wid * 32;
    const int hh   = lane >> 4;          // half-wave: 0 or 1
    const int arow = lane & 15;          // A-matrix M row owned by this lane
    const int ksel = hh * 2;             // lanes 16-31 hold K+2, K+3
    const int col0 = n0 + (lane & 15);
    const int col1 = col0 + 16;

    v8f acc0 = {};
    v8f acc1 = {};
    const float* lA = atile + arow * LDS_PITCH + ksel;

    #pragma unroll 8
    for (int kk = 0; kk < IN_FEAT; kk += 4) {
        // A 16x4 (fp32): lane L<16 -> {A[L][kk], A[L][kk+1]}, lane L>=16 -> {A[L-16][kk+2], A[L-16][kk+3]}
        v2f a = *(const v2f*)(lA + kk);          // 8B-aligned ds_load_b64

        // B 4x16 (fp32), mirrored layout: VGPR0 = rows K/K+2, VGPR1 = rows K+1/K+3
        const int kr = kk + ksel;
        v2f b0, b1;
        b0.x = w_mat[(size_t)kr * UNITS + col0];
        b0.y = w_mat[(size_t)(kr + 1) * UNITS + col0];
        b1.x = w_mat[(size_t)kr * UNITS + col1];
        b1.y = w_mat[(size_t)(kr + 1) * UNITS + col1];

        acc0 = __builtin_amdgcn_wmma_f32_16x16x4_f32(false, a, false, b0, (short)0, acc0, false, false);
        acc1 = __builtin_amdgcn_wmma_f32_16x16x4_f32(false, a, false, b1, (short)0, acc1, false, false);
    }

    // D layout: VGPR r -> M = r (lanes 0-15) / r+8 (lanes 16-31), N = lane&15.
    float* hp = h + (size_t)(m0 + hh * 8) * UNITS + col0;
    #pragma unroll
    for (int r = 0; r < 8; ++r) {
        hp[(size_t)r * UNITS]      = acc0[r];
        hp[(size_t)r * UNITS + 16] = acc1[r];
    }
}

// ---------------------------------------------------------------------------
// Zero-init of out (harness poisons d_out with 0xAA).
// ---------------------------------------------------------------------------
__global__ void gcn_zero_f4(float4* __restrict__ out, int n4)
{
    int i = blockIdx.x * blockDim.x + threadIdx.x;
    if (i < n4) out[i] = make_float4(0.f, 0.f, 0.f, 0.f);
}

// ---------------------------------------------------------------------------
// Edge scatter: one wave32 per edge. Each lane handles 8 floats (2 x float4
// gather from h[col], 8 global_atomic_add_f32 into out[row]).
// ---------------------------------------------------------------------------
__global__ void __launch_bounds__(256) gcn_spmm_scatter(
    const float* __restrict__ h, const float* __restrict__ vals,
    const int* __restrict__ rows, const int* __restrict__ cols,
    float* __restrict__ out, int n_edges)
{
    const int e = (int)((blockIdx.x * (unsigned)blockDim.x + threadIdx.x) >> 5);
    if (e >= n_edges) return;
    const int lane = threadIdx.x & 31;

    const int   c = cols[e];
    const int   r = rows[e];
    const float v = vals[e];

    const float4* hv = (const float4*)(h + (size_t)c * UNITS);
    float*        ov = out + (size_t)r * UNITS;

    float4 p0 = hv[lane];        // floats [lane*4,   lane*4+4)
    float4 p1 = hv[lane + 32];   // floats [128+lane*4, 128+lane*4+4)

    const int b0 = lane * 4;
    const int b1 = b0 + 128;
    atomicAdd(ov + b0 + 0, v * p0.x);
    atomicAdd(ov + b0 + 1, v * p0.y);
    atomicAdd(ov + b0 + 2, v * p0.z);
    atomicAdd(ov + b0 + 3, v * p0.w);
    atomicAdd(ov + b1 + 0, v * p1.x);
    atomicAdd(ov + b1 + 1, v * p1.y);
    atomicAdd(ov + b1 + 2, v * p1.z);
    atomicAdd(ov + b1 + 3, v * p1.w);
}

// ---------------------------------------------------------------------------
// ReLU pass over out (float4 for bandwidth).
// ---------------------------------------------------------------------------
__global__ void gcn_relu_f4(float4* __restrict__ out, int n4)
{
    int i = blockIdx.x * blockDim.x + threadIdx.x;
    if (i < n4) {
        float4 t = out[i];
        t.x = t.x > 0.f ? t.x : 0.f;
        t.y = t.y > 0.f ? t.y : 0.f;
        t.z = t.z > 0.f ? t.z : 0.f;
        t.w = t.w > 0.f ? t.w : 0.f;
        out[i] = t;
    }
}

extern "C" void kernel_launch(void* const* d_in, const int* in_sizes, int n_in,
                              void* d_out, int out_size, void* d_ws, size_t ws_size,
                              hipStream_t stream)
{
    const float* x    = (const float*)d_in[0];
    const float* w    = (const float*)d_in[1];
    const float* vals = (const float*)d_in[2];
    const int*   rows = (const int*)d_in[3];
    const int*   cols = (const int*)d_in[4];
    float*       out  = (float*)d_out;
    float*       h    = (float*)d_ws;   // 50000*256*4 = 51.2 MB scratch

    const int n_nodes = in_sizes[0] / IN_FEAT;   // 50000 (multiple of 16)
    const int n_edges = in_sizes[2];             // 800000

    const int n4 = n_nodes * UNITS / 4;

    // 1) zero the accumulation buffer
    gcn_zero_f4<<<(n4 + 255) / 256, 256, 0, stream>>>((float4*)out, n4);

    // 2) dense projection h = x @ w  (WMMA f32, one block per 16-row tile)
    gcn_gemm_wmma_f32<<<n_nodes / 16, 256, 0, stream>>>(x, w, h);

    // 3) edge gather + atomic scatter (one wave per edge, 8 waves/block)
    gcn_spmm_scatter<<<(n_edges + 7) / 8, 256, 0, stream>>>(h, vals, rows, cols, out, n_edges);

    // 4) ReLU
    gcn_relu_f4<<<(n4 + 255) / 256, 256, 0, stream>>>((float4*)out, n4);
}